// MultiHeadMesa_1314259993119
// MI455X (gfx1250) — compile-verified
//
#include <hip/hip_runtime.h>

typedef __attribute__((ext_vector_type(2))) float v2f;
typedef __attribute__((ext_vector_type(8))) float v8f;
typedef __attribute__((ext_vector_type(4))) unsigned int u32x4;
typedef __attribute__((ext_vector_type(8))) int i32x8;
typedef __attribute__((ext_vector_type(4))) int i32x4;

#define T_DIM 1024
#define E_DIM 512
#define H_DIM 8
#define D_DIM 64
#define B_DIM 2
#define NROW (B_DIM * T_DIM) /* 2048 */

static __device__ __forceinline__ v8f wmma_f32_16x16x4(v2f a, v2f b, v8f c) {
  // 8-arg form: (neg_a, A, neg_b, B, c_mod, C, reuse_a, reuse_b)
  return __builtin_amdgcn_wmma_f32_16x16x4_f32(false, a, false, b, (short)0, c,
                                               false, false);
}

// Low 32 bits of a generic pointer to LDS == LDS byte offset (ISA 10.2:
// LDS uses the low 32 address bits).
static __device__ __forceinline__ unsigned lds_off(const void* p) {
  return (unsigned)(unsigned long long)(size_t)p;
}

// ---------------------------------------------------------------------------
// TDM descriptor group 1 for a 2-D fp32 tile load (invariant per tile shape):
//   data_size=2 (4B) [17:16], pad_enable [20], pad_interval [24:22]
//   (code c -> pad every 2^(c+1) dwords), pad_amount [31:25] (code c -> c+1
//   dwords), tensor_dim0 = tile width (no OOB), tensor_dim1 = tile height,
//   tensor_dim0_stride = row stride.  workgroup_mask=0 (not in a cluster).
// ---------------------------------------------------------------------------
static __device__ __forceinline__ i32x8 tdm_g1_2d(unsigned tile_w,
                                                  unsigned tile_h,
                                                  unsigned row_stride_elems,
                                                  unsigned pad_interval_code,
                                                  unsigned pad_amount_code) {
  i32x8 g1;
  g1[0] = (int)((2u << 16) | (1u << 20) | (pad_interval_code << 22) |
                (pad_amount_code << 25));
  g1[1] = (int)((tile_w & 0xFFFFu) << 16);  // tensor_dim0 lo16
  g1[2] = (int)(((tile_w >> 16) & 0xFFFFu) | ((tile_h & 0xFFFFu) << 16));
  g1[3] = (int)(((tile_h >> 16) & 0xFFFFu) | ((tile_w & 0xFFFFu) << 16));
  g1[4] = (int)(tile_h & 0xFFFFu);          // tile_dim1; tile_dim2=0
  g1[5] = (int)row_stride_elems;            // tensor_dim0_stride lo32
  g1[6] = 0;
  g1[7] = 0;
  return g1;
}

// Issue a TDM tile load (TENSORcnt-tracked).  Only group 0 varies per issue.
static __device__ __forceinline__ void tdm_issue(const float* gaddr,
                                                 unsigned lds_byte_off,
                                                 i32x8 g1) {
  const unsigned long long ga = (unsigned long long)(size_t)gaddr;
  u32x4 g0;
  g0[0] = 1u;                                        // count=1, user mode
  g0[1] = lds_byte_off;                              // lds_addr
  g0[2] = (unsigned)(ga & 0xFFFFFFFFu);              // global_addr[31:0]
  g0[3] = (unsigned)((ga >> 32) & 0x01FFFFFFu) | (2u << 30);  // hi | type=2
  i32x4 g2 = {0, 0, 0, 0};  // 2-D tile: dims 2+ unused
  i32x4 g3 = {0, 0, 0, 0};
#if __clang_major__ >= 23
  i32x8 g4 = {0, 0, 0, 0, 0, 0, 0, 0};
  __builtin_amdgcn_tensor_load_to_lds(g0, g1, g2, g3, g4, 0);
#else
  __builtin_amdgcn_tensor_load_to_lds(g0, g1, g2, g3, 0);
#endif
}

// ---------------------------------------------------------------------------
// Kernel 1/4: WMMA GEMM, one wave per 64x64 output macro-tile.
//   A[2048 x 512] @ W[512 x 512].  K-slabs of 32 staged into double-buffered
//   LDS by TDM (A: 64x32 rows padded to 34 dwords; B: 32x64 rows padded to 66).
// __launch_bounds__(32,1): register-fat single-wave workgroup -> no spills
// (16 v8f accumulators = 128 VGPRs must stay resident).
// scatter_qkv=1: write into [B,H,T,d] head layout; else plain row-major.
// ---------------------------------------------------------------------------
#define AST 34  /* A LDS row stride (dwords): pad 32+2, even -> b64 loads */
#define BST 66  /* B LDS row stride (dwords): pad 64+2 */

__global__ void __launch_bounds__(32, 1)
gemm64_tdm_kernel(const float* __restrict__ Amat, const float* __restrict__ W0,
                  const float* __restrict__ W1, const float* __restrict__ W2,
                  float* __restrict__ D0, float* __restrict__ D1,
                  float* __restrict__ D2, int scatter_qkv) {
  __shared__ __align__(16) float lA[2][64 * AST];
  __shared__ __align__(16) float lB[2][32 * BST];

  const int which = blockIdx.z;
  const float* W = (which == 0) ? W0 : ((which == 1) ? W1 : W2);
  float* D = (which == 0) ? D0 : ((which == 1) ? D1 : D2);

  const int r0 = blockIdx.x * 64;
  const int c0 = blockIdx.y * 64;
  const int lane = threadIdx.x;
  const int row = lane & 15;
  const int kb = (lane >> 4) * 2;
  const int col = lane & 15;
  const int mhi = (lane >> 4) * 8;

  const i32x8 g1A = tdm_g1_2d(32, 64, E_DIM, /*every 32 dw*/ 4, /*2 dw*/ 1);
  const i32x8 g1B = tdm_g1_2d(64, 32, E_DIM, /*every 64 dw*/ 5, /*2 dw*/ 1);

  v8f acc[4][4] = {};

  // Prologue: slab 0 in flight.
  tdm_issue(Amat + (size_t)r0 * E_DIM, lds_off(&lA[0][0]), g1A);
  tdm_issue(W + c0, lds_off(&lB[0][0]), g1B);

  const int NSLAB = E_DIM / 32;  // 16
  for (int s = 0; s < NSLAB; ++s) {
    const int cur = s & 1;
    if (s + 1 < NSLAB) {
      const int k1 = (s + 1) * 32;
      tdm_issue(Amat + (size_t)r0 * E_DIM + k1, lds_off(&lA[cur ^ 1][0]), g1A);
      tdm_issue(W + (size_t)k1 * E_DIM + c0, lds_off(&lB[cur ^ 1][0]), g1B);
      __builtin_amdgcn_s_wait_tensorcnt(2);  // slab s resident, s+1 in flight
    } else {
      __builtin_amdgcn_s_wait_tensorcnt(0);
    }
    asm volatile("" ::: "memory");

    const float* a_ = &lA[cur][0];
    const float* b_ = &lB[cur][0];
#pragma unroll
    for (int k = 0; k < 32; k += 4) {
      v2f af[4], bf[4];
#pragma unroll
      for (int m = 0; m < 4; ++m) {
        const float* ap = a_ + (m * 16 + row) * AST + k + kb;  // 8B aligned
        af[m].x = ap[0];
        af[m].y = ap[1];
      }
#pragma unroll
      for (int n = 0; n < 4; ++n) {
        bf[n].x = b_[(k + kb) * BST + n * 16 + col];
        bf[n].y = b_[(k + kb + 1) * BST + n * 16 + col];
      }
#pragma unroll
      for (int m = 0; m < 4; ++m)
#pragma unroll
        for (int n = 0; n < 4; ++n)
          acc[m][n] = wmma_f32_16x16x4(af[m], bf[n], acc[m][n]);
    }
    asm volatile("" ::: "memory");
  }

  // Store 64x64 macro-tile.
#pragma unroll
  for (int m = 0; m < 4; ++m) {
#pragma unroll
    for (int n = 0; n < 4; ++n) {
#pragma unroll
      for (int i = 0; i < 8; ++i) {
        const int R = r0 + m * 16 + i + mhi;
        const int ncol = c0 + n * 16 + col;
        if (scatter_qkv) {
          const int b_i = R >> 10;
          const int t = R & (T_DIM - 1);
          const int h = ncol >> 6;
          const int di = ncol & 63;
          D[(((size_t)(b_i * H_DIM + h)) * T_DIM + t) * D_DIM + di] =
              acc[m][n][i];
        } else {
          D[(size_t)R * E_DIM + ncol] = acc[m][n][i];
        }
      }
    }
  }
}

// ---------------------------------------------------------------------------
// Kernel 2: Sherman-Morrison scan.  One 256-thread block per (b,h) chain.
// inv (64x64, symmetric) lives in LDS with stride-65 padding.
//   Hx = inv @ x_t ; Hq = inv @ q_t (same LDS pass)
//   denom = 1 + x.Hx ; inv -= outer(Hx,Hx)/denom
//   pred  = Hq - Hx * (Hx.q)/denom        (== new_inv @ q_t)
// ---------------------------------------------------------------------------
__global__ void mesa_scan_kernel(const float* __restrict__ vbuf,
                                 const float* __restrict__ qbuf,
                                 const float* __restrict__ lambdas,
                                 float* __restrict__ preds) {
  __shared__ float inv[D_DIM * 65];
  __shared__ float xv[D_DIM], qv[D_DIM];
  __shared__ float Hx[D_DIM], Hq[D_DIM];
  __shared__ float part[D_DIM * 4], partq[D_DIM * 4];
  __shared__ float prod[D_DIM], prodq[D_DIM];
  __shared__ float red[16], redq[16];
  __shared__ float sc[2];  // {1/denom, Hx.q}

  const int bh = blockIdx.x;
  const int h = bh & 7;
  const int tid = threadIdx.x;
  const float lam = lambdas[h];

  const float* vp = vbuf + (size_t)bh * T_DIM * D_DIM;
  const float* qp = qbuf + (size_t)bh * T_DIM * D_DIM;
  float* pp = preds + (size_t)bh * T_DIM * D_DIM;

  for (int e = tid; e < D_DIM * D_DIM; e += 256) {
    const int i = e >> 6, j = e & 63;
    inv[i * 65 + j] = (i == j) ? lam : 0.0f;
  }
  __syncthreads();

  const int row = tid >> 2;
  const int seg = (tid & 3) * 16;

  for (int t = 0; t < T_DIM; ++t) {
    if (tid < D_DIM) {
      xv[tid] = vp[t * D_DIM + tid];
      qv[tid] = qp[t * D_DIM + tid];
    }
    __syncthreads();

    float px = 0.f, pq = 0.f;
#pragma unroll
    for (int jj = 0; jj < 16; ++jj) {
      const float m = inv[row * 65 + seg + jj];
      px += m * xv[seg + jj];
      pq += m * qv[seg + jj];
    }
    part[row * 4 + (tid & 3)] = px;
    partq[row * 4 + (tid & 3)] = pq;
    __syncthreads();

    if (tid < D_DIM) {
      const float hx = part[tid * 4] + part[tid * 4 + 1] + part[tid * 4 + 2] +
                       part[tid * 4 + 3];
      const float hq = partq[tid * 4] + partq[tid * 4 + 1] +
                       partq[tid * 4 + 2] + partq[tid * 4 + 3];
      Hx[tid] = hx;
      Hq[tid] = hq;
      prod[tid] = hx * xv[tid];
      prodq[tid] = hx * qv[tid];
    }
    __syncthreads();

    if (tid < 16) {
      red[tid] = prod[tid * 4] + prod[tid * 4 + 1] + prod[tid * 4 + 2] +
                 prod[tid * 4 + 3];
      redq[tid] = prodq[tid * 4] + prodq[tid * 4 + 1] + prodq[tid * 4 + 2] +
                  prodq[tid * 4 + 3];
    }
    __syncthreads();

    if (tid == 0) {
      float s = 0.f, sq = 0.f;
#pragma unroll
      for (int i = 0; i < 16; ++i) {
        s += red[i];
        sq += redq[i];
      }
      sc[0] = 1.0f / (1.0f + s);
      sc[1] = sq;
    }
    __syncthreads();

    const float rd = sc[0];
    const float hxr = Hx[row];
#pragma unroll
    for (int jj = 0; jj < 16; ++jj) {
      inv[row * 65 + seg + jj] -= hxr * Hx[seg + jj] * rd;
    }
    if (tid < D_DIM) {
      pp[t * D_DIM + tid] = Hq[tid] - Hx[tid] * sc[1] * rd;
    }
    __syncthreads();
  }
}

// ---------------------------------------------------------------------------
// Kernel 3: causal linear attention: O = tril(P @ V^T) @ K, per (b,h).
// One wave per 16-row block; S round-trips through LDS to become an A-frag.
// Output written directly in the reference's [B,T,d*H] reshape order.
// ---------------------------------------------------------------------------
__global__ void mesa_attn_kernel(const float* __restrict__ preds,
                                 const float* __restrict__ vbuf,
                                 const float* __restrict__ kbuf,
                                 float* __restrict__ mesa) {
  __shared__ float sT[16 * 17];
  const int r0 = blockIdx.x * 16;
  const int bh = blockIdx.y;
  const int b_ = bh >> 3, h = bh & 7;
  const float* P = preds + (size_t)bh * T_DIM * D_DIM;
  const float* V = vbuf + (size_t)bh * T_DIM * D_DIM;
  const float* K = kbuf + (size_t)bh * T_DIM * D_DIM;

  const int lane = threadIdx.x;
  const int row = lane & 15;
  const int kb = (lane >> 4) * 2;
  const int col = lane & 15;
  const int mhi = (lane >> 4) * 8;

  v8f o0 = {}, o1 = {}, o2 = {}, o3 = {};

  for (int j0 = 0; j0 <= r0; j0 += 16) {
    // S = P[r0:+16, :] @ V[j0:+16, :]^T   (K = 64)
    v8f s = {};
#pragma unroll
    for (int k0 = 0; k0 < D_DIM; k0 += 4) {
      v2f a, bb;
      const float* ap = P + (size_t)(r0 + row) * D_DIM + k0 + kb;
      a.x = ap[0];
      a.y = ap[1];
      const float* bp = V + (size_t)(j0 + col) * D_DIM + k0 + kb;
      bb.x = bp[0];
      bb.y = bp[1];
      s = wmma_f32_16x16x4(a, bb, s);
    }
    if (j0 == r0) {  // diagonal block: zero strict upper triangle
#pragma unroll
      for (int i = 0; i < 8; ++i) {
        if ((i + mhi) < col) s[i] = 0.0f;
      }
    }
#pragma unroll
    for (int i = 0; i < 8; ++i) sT[(i + mhi) * 17 + col] = s[i];
    __syncthreads();

    // O += S @ K[j0:+16, :]   (K = 16, four 16-wide d chunks)
#pragma unroll
    for (int k0 = 0; k0 < 16; k0 += 4) {
      v2f a;
      a.x = sT[row * 17 + k0 + kb];
      a.y = sT[row * 17 + k0 + kb + 1];
      const float* kp0 = K + (size_t)(j0 + k0 + kb) * D_DIM;
      const float* kp1 = K + (size_t)(j0 + k0 + kb + 1) * D_DIM;
      v2f b0, b1, b2, b3;
      b0.x = kp0[0 + col];  b0.y = kp1[0 + col];
      b1.x = kp0[16 + col]; b1.y = kp1[16 + col];
      b2.x = kp0[32 + col]; b2.y = kp1[32 + col];
      b3.x = kp0[48 + col]; b3.y = kp1[48 + col];
      o0 = wmma_f32_16x16x4(a, b0, o0);
      o1 = wmma_f32_16x16x4(a, b1, o1);
      o2 = wmma_f32_16x16x4(a, b2, o2);
      o3 = wmma_f32_16x16x4(a, b3, o3);
    }
    __syncthreads();
  }

  // mesa[b, t, di*H + h]  (reference reshape combines (d,H) in that order)
#pragma unroll
  for (int i = 0; i < 8; ++i) {
    const int t = r0 + i + mhi;
    const size_t base = ((size_t)(b_ * T_DIM + t)) * (D_DIM * H_DIM) + h;
    mesa[base + (size_t)(0 * 16 + col) * H_DIM] = o0[i];
    mesa[base + (size_t)(1 * 16 + col) * H_DIM] = o1[i];
    mesa[base + (size_t)(2 * 16 + col) * H_DIM] = o2[i];
    mesa[base + (size_t)(3 * 16 + col) * H_DIM] = o3[i];
  }
}

extern "C" void kernel_launch(void* const* d_in, const int* in_sizes, int n_in,
                              void* d_out, int out_size, void* d_ws,
                              size_t ws_size, hipStream_t stream) {
  (void)in_sizes; (void)n_in; (void)out_size; (void)ws_size;
  const float* x = (const float*)d_in[0];
  const float* Wq = (const float*)d_in[1];
  const float* Wk = (const float*)d_in[2];
  const float* Wv = (const float*)d_in[3];
  const float* Wo = (const float*)d_in[4];
  const float* lambdas = (const float*)d_in[5];

  float* ws = (float*)d_ws;
  const size_t SZ = (size_t)NROW * E_DIM;  // 1,048,576 floats (4 MB)
  float* qbuf = ws;             // [B,H,T,d]
  float* kbuf = ws + 1 * SZ;    // [B,H,T,d]
  float* vbuf = ws + 2 * SZ;    // [B,H,T,d]
  float* preds = ws + 3 * SZ;   // [B,H,T,d]
  float* mesa = ws + 4 * SZ;    // [B,T, d*H]

  // QKV projections (TDM-staged WMMA GEMM, head-scatter output).
  gemm64_tdm_kernel<<<dim3(NROW / 64, E_DIM / 64, 3), 32, 0, stream>>>(
      x, Wq, Wk, Wv, qbuf, kbuf, vbuf, 1);
  // Sequential Sherman-Morrison scan (latency critical path, 16 chains).
  mesa_scan_kernel<<<dim3(B_DIM * H_DIM), 256, 0, stream>>>(vbuf, qbuf,
                                                            lambdas, preds);
  // Causal linear attention.
  mesa_attn_kernel<<<dim3(T_DIM / 16, B_DIM * H_DIM), 32, 0, stream>>>(
      preds, vbuf, kbuf, mesa);
  // Output projection (plain row-major store).
  gemm64_tdm_kernel<<<dim3(NROW / 64, E_DIM / 64, 1), 32, 0, stream>>>(
      mesa, Wo, Wo, Wo, (float*)d_out, nullptr, nullptr, 0);
}